// Ffjord_11854109736969
// MI455X (gfx1250) — compile-verified
//
#include <hip/hip_runtime.h>
#include <hip/hip_bf16.h>
#include <math.h>

// ---------------------------------------------------------------------------
// FFJORD reverse CNF for MI455X (gfx1250, wave32, WMMA).
//
// Stage 1 (ffjord_hyper): hyper(t) for the 5 distinct t values {0,.5,1,1.5,2}
//   -> table in d_ws: per t (256 floats): W[96] | U[96] | Bb[32] | WdotU[32]
// Stage 2 (ffjord_main): 16 samples per wave. Per func eval:
//   D' = W_chunk[16x4] x Z^T[4x16] via V_WMMA_F32_16X16X4_F32 (2 chunks for
//   WIDTH=32), then branchless tanh + U/trace accumulation in-lane, one
//   shfl_xor(16) cross-half reduction, RK4 combine.
// ---------------------------------------------------------------------------

typedef __attribute__((ext_vector_type(2))) float v2f;
typedef __attribute__((ext_vector_type(8))) float v8f;

#define NT      5    // distinct t values
#define TSTRIDE 256

// ---- branchless transcendentals (no EXEC-masked OCML paths) ---------------
__device__ __forceinline__ float tanh_fast(float x) {
#if __has_builtin(__builtin_amdgcn_tanhf)
    return __builtin_amdgcn_tanhf(x);            // v_tanh_f32 (gfx1250 TRANS)
#elif __has_builtin(__builtin_amdgcn_tanh_f32)
    return __builtin_amdgcn_tanh_f32(x);
#else
    // tanh(x) = sign(x) * (1-e)/(1+e), e = exp(-2|x|) = exp2(-2*log2e*|x|)
    const float ax = __builtin_fabsf(x);
    const float e  = __builtin_amdgcn_exp2f(ax * -2.885390081777927f);
    const float r  = (1.0f - e) * __builtin_amdgcn_rcpf(1.0f + e);
    return __builtin_copysignf(r, x);
#endif
}

__device__ __forceinline__ float sigmoid_fast(float x) {
    // 1/(1+exp(-x)) = 1/(1+exp2(-log2e*x))
    const float e = __builtin_amdgcn_exp2f(x * -1.4426950408889634f);
    return __builtin_amdgcn_rcpf(1.0f + e);
}

// ---------------- Stage 1: hypernetwork table ------------------------------
__global__ __launch_bounds__(32) void ffjord_hyper(
    const float* __restrict__ w1, const float* __restrict__ b1,
    const float* __restrict__ w2, const float* __restrict__ b2,
    const float* __restrict__ wW, const float* __restrict__ bW,
    const float* __restrict__ wU, const float* __restrict__ bU,
    const float* __restrict__ wG, const float* __restrict__ bG,
    const float* __restrict__ wB, const float* __restrict__ bB,
    float* __restrict__ tab)
{
    __shared__ float h1[32];
    __shared__ float h2[32];
    const int w = threadIdx.x;                // 0..31
    const float t = 0.5f * (float)blockIdx.x; // t index k -> t = 0.5*k

    h1[w] = tanh_fast(t * w1[w] + b1[w]);
    __syncthreads();

    float acc = b2[w];
    #pragma unroll 8
    for (int j = 0; j < 32; ++j) acc += w2[w * 32 + j] * h1[j];
    h2[w] = tanh_fast(acc);
    __syncthreads();

    float* base = tab + (size_t)blockIdx.x * TSTRIDE;
    float Wr[3], Ur[3];
    #pragma unroll
    for (int d = 0; d < 3; ++d) {
        const int i = w * 3 + d;
        float aw = bW[i], au = bU[i], ag = bG[i];
        #pragma unroll 8
        for (int j = 0; j < 32; ++j) {
            const float h = h2[j];
            aw += wW[i * 32 + j] * h;
            au += wU[i * 32 + j] * h;
            ag += wG[i * 32 + j] * h;
        }
        const float u = au * sigmoid_fast(ag);
        Wr[d] = aw; Ur[d] = u;
        base[i]      = aw;   // W   at [0,96)
        base[96 + i] = u;    // U   at [96,192)
    }
    float ab = bB[w];
    #pragma unroll 8
    for (int j = 0; j < 32; ++j) ab += wB[w * 32 + j] * h2[j];
    base[192 + w] = ab;                                        // Bb
    base[224 + w] = Wr[0]*Ur[0] + Wr[1]*Ur[1] + Wr[2]*Ur[2];   // W.U
}

// ---------------- func(z, t): one dynamics evaluation ----------------------
// P points at the 256-float table for this t (in LDS).
// Lane roles: n = lane&15 owns sample n of the tile; half = lane>>4.
__device__ __forceinline__ void func_eval(const float* __restrict__ P,
                                          float x0, float x1, float x2,
                                          int half, int n,
                                          float dz[3], float* ell)
{
    // B operand: Z^T [4 x 16] (K x N), K=3 padded to 4.
    // lane = N; lanes 0-15 hold K=0(V0),K=1(V1); lanes 16-31 hold K=2(V0),K=3(V1).
    v2f bop;
    bop.x = half ? x2 : x0;
    bop.y = half ? 0.0f : x1;

    float f0 = 0.f, f1 = 0.f, f2 = 0.f, tr = 0.f;

    #pragma unroll
    for (int c = 0; c < 2; ++c) {               // width chunks 0-15 / 16-31
        const int m = c * 16 + n;               // A-matrix row (width) for lane
        // Unconditional b96 load of the W row, then register selects
        // (avoids exec-predicated ds_load).
        const float w0 = P[m * 3 + 0];
        const float w1v = P[m * 3 + 1];
        const float w2v = P[m * 3 + 2];
        // A operand: W_chunk [16 x 4] (M x K); lane = M,
        // lanes 0-15: K=0(V0),K=1(V1); lanes 16-31: K=2(V0),K=3(V1 pad=0)
        v2f aop;
        aop.x = half ? w2v : w0;
        aop.y = half ? 0.0f : w1v;

        v8f acc = {};
        acc = __builtin_amdgcn_wmma_f32_16x16x4_f32(
                  /*neg_a=*/false, aop, /*neg_b=*/false, bop,
                  /*c_mod=*/(short)0, acc, /*reuse_a=*/false, /*reuse_b=*/false);

        const int wbase = c * 16 + half * 8;    // widths held by this lane
        #pragma unroll
        for (int r = 0; r < 8; ++r) {
            const int w = wbase + r;
            const float h = tanh_fast(acc[r] + P[192 + w]);
            f0 += h * P[96 + w * 3 + 0];
            f1 += h * P[96 + w * 3 + 1];
            f2 += h * P[96 + w * 3 + 2];
            tr += (1.0f - h * h) * P[224 + w];
        }
    }
    // combine the two lane-halves (each summed 16 of the 32 widths)
    f0 += __shfl_xor(f0, 16);
    f1 += __shfl_xor(f1, 16);
    f2 += __shfl_xor(f2, 16);
    tr += __shfl_xor(tr, 16);

    const float inv = 1.0f / 32.0f;
    dz[0] = f0 * inv; dz[1] = f1 * inv; dz[2] = f2 * inv;
    *ell  = -tr * inv;
}

// ---------------- Stage 2: main integration kernel -------------------------
__global__ __launch_bounds__(256) void ffjord_main(
    const float* __restrict__ z1, const float* __restrict__ tab,
    float* __restrict__ out, int Bn)
{
    __shared__ float P[NT * TSTRIDE];
    for (int i = threadIdx.x; i < NT * TSTRIDE; i += blockDim.x) P[i] = tab[i];
    __syncthreads();

    const int lane = threadIdx.x & 31;
    const int half = lane >> 4;
    const int n    = lane & 15;
    const int wave = blockIdx.x * (blockDim.x >> 5) + (threadIdx.x >> 5);
    const int numTiles = Bn >> 4;               // B divisible by 16 (1e6/16)
    if (wave >= numTiles) return;               // whole-wave exit: EXEC stays full

    const int s = wave * 16 + n;                // 32-bit indexing: 3B < 2^31
    float zc0 = z1[s * 3 + 0];
    float zc1 = z1[s * 3 + 1];
    float zc2 = z1[s * 3 + 2];
    float logdet = 0.0f;

    const float dt = -1.0f;                     // both RK4 steps use dt = -1
    #pragma unroll
    for (int step = 0; step < 2; ++step) {
        // t-table indices: step0 -> t1=2: (4,3,3,2); step1 -> t1=1: (2,1,1,0)
        const int tA = 4 - 2 * step, tB = tA - 1, tC = tA - 2;
        float k1[3], k2[3], k3[3], k4[3], l1, l2, l3, l4;

        func_eval(P + tA * TSTRIDE, zc0, zc1, zc2, half, n, k1, &l1);
        func_eval(P + tB * TSTRIDE,
                  zc0 + 0.5f * dt * k1[0], zc1 + 0.5f * dt * k1[1],
                  zc2 + 0.5f * dt * k1[2], half, n, k2, &l2);
        func_eval(P + tB * TSTRIDE,
                  zc0 + 0.5f * dt * k2[0], zc1 + 0.5f * dt * k2[1],
                  zc2 + 0.5f * dt * k2[2], half, n, k3, &l3);
        func_eval(P + tC * TSTRIDE,
                  zc0 + dt * k3[0], zc1 + dt * k3[1],
                  zc2 + dt * k3[2], half, n, k4, &l4);

        const float c6 = dt * (1.0f / 6.0f);
        zc0 += c6 * (k1[0] + 2.f * k2[0] + 2.f * k3[0] + k4[0]);
        zc1 += c6 * (k1[1] + 2.f * k2[1] + 2.f * k3[1] + k4[1]);
        zc2 += c6 * (k1[2] + 2.f * k2[2] + 2.f * k3[2] + k4[2]);
        logdet -= c6 * (l1 + 2.f * l2 + 2.f * l3 + l4);
    }

    if (half == 0) {                            // lanes 0-15 own the 16 samples
        out[s * 3 + 0] = zc0;
        out[s * 3 + 1] = zc1;
        out[s * 3 + 2] = zc2;
        out[Bn * 3 + s] = logdet;
    }
}

// ---------------------------------------------------------------------------
extern "C" void kernel_launch(void* const* d_in, const int* in_sizes, int n_in,
                              void* d_out, int out_size, void* d_ws, size_t ws_size,
                              hipStream_t stream)
{
    const float* z1 = (const float*)d_in[0];
    const float* w1 = (const float*)d_in[1];
    const float* b1 = (const float*)d_in[2];
    const float* w2 = (const float*)d_in[3];
    const float* b2 = (const float*)d_in[4];
    const float* wW = (const float*)d_in[5];
    const float* bW = (const float*)d_in[6];
    const float* wU = (const float*)d_in[7];
    const float* bU = (const float*)d_in[8];
    const float* wG = (const float*)d_in[9];
    const float* bG = (const float*)d_in[10];
    const float* wB = (const float*)d_in[11];
    const float* bB = (const float*)d_in[12];
    float* out = (float*)d_out;
    float* tab = (float*)d_ws;                  // 5*256 floats = 5 KB

    const int Bn = in_sizes[0] / 3;

    ffjord_hyper<<<dim3(NT), dim3(32), 0, stream>>>(
        w1, b1, w2, b2, wW, bW, wU, bU, wG, bG, wB, bB, tab);

    const int numTiles = Bn / 16;               // 62500 for B=1e6
    const int wavesPerBlock = 256 / 32;         // 8 tiles per block
    const int blocks = (numTiles + wavesPerBlock - 1) / wavesPerBlock;
    ffjord_main<<<dim3(blocks), dim3(256), 0, stream>>>(z1, tab, out, Bn);
}